// DistributedMemory_51238959841356
// MI455X (gfx1250) — compile-verified
//
#include <hip/hip_runtime.h>

#define VEC_DIM 128
#define NWORDS  100000
#define CCTX    8
#define SSAMP   10
#define WAVES_PER_BLOCK 8

typedef float v2f __attribute__((ext_vector_type(2)));
typedef float v8f __attribute__((ext_vector_type(8)));

__global__ __launch_bounds__(256) void pvdm_wmma_kernel(
    const int*   __restrict__ doc_ids,      // [B]
    const int*   __restrict__ context_ids,  // [B, C]
    const int*   __restrict__ sample_ids,   // [B, S]
    const float* __restrict__ P,            // [N_DOCS, 128]
    const float* __restrict__ Wm,           // [N_WORDS, 128]
    const float* __restrict__ Out,          // [128, N_WORDS]
    float*       __restrict__ res,          // [B, S]
    int Btot)
{
    __shared__ __align__(16) float ldsf[WAVES_PER_BLOCK * VEC_DIM];

    const int lane = threadIdx.x & 31;
    const int wave = threadIdx.x >> 5;
    const int b    = blockIdx.x * WAVES_PER_BLOCK + wave;
    if (b >= Btot) return;   // wave-uniform; grid is sized exactly so WMMA waves keep EXEC all-1s

    // ---------------- Phase 1: inputs[b,:] = P[doc] + mean(Wm[ctx]) -> LDS ----------------
    // Coalesced: each lane owns 4 consecutive dims (float4), 32 lanes cover the 128-dim row.
    const int doc = doc_ids[b];
    float4 p = ((const float4*)(P + (size_t)doc * VEC_DIM))[lane];

    float4 s = make_float4(0.f, 0.f, 0.f, 0.f);
    #pragma unroll
    for (int c = 0; c < CCTX; ++c) {
        const int w = context_ids[b * CCTX + c];
        float4 t = ((const float4*)(Wm + (size_t)w * VEC_DIM))[lane];
        s.x += t.x; s.y += t.y; s.z += t.z; s.w += t.w;
    }
    float4 inp;
    inp.x = p.x + 0.125f * s.x;
    inp.y = p.y + 0.125f * s.y;
    inp.z = p.z + 0.125f * s.z;
    inp.w = p.w + 0.125f * s.w;

    ((float4*)&ldsf[wave * VEC_DIM])[lane] = inp;
    // Same-wave LDS RAW: LDS pipe is in-order, but be explicit.
    asm volatile("s_wait_dscnt 0x0" ::: "memory");

    // ---------------- Phase 2: D[m,n] += A[m,k] * Bb[k,n], 32 chained K=4 steps ----------------
    // A (16x4 f32): lane m in 0..15 -> A[m, 0],A[m, 1]; lane 16+m -> A[m, 2],A[m, 3]
    // A[m,k] = Out[(kc+k)*NWORDS + col_m]  (gathered column of the output matrix)
    // Bb (4x16 f32): broadcast of inputs vector across all 16 columns:
    //   lanes 0..15: K=kc+0 (V0), K=kc+1 (V1); lanes 16..31: K=kc+2 (V0), K=kc+3 (V1)
    const int m    = lane & 15;
    const int koff = (lane >> 4) << 1;                       // 0 for lanes 0-15, 2 for lanes 16-31
    const int srow = (m < SSAMP) ? m : (SSAMP - 1);          // pad rows 10..15 with a dup (cache hit)
    const int col  = sample_ids[b * SSAMP + srow];

    const float* aptr  = Out + (size_t)koff * NWORDS + (size_t)col;
    const float* bbase = &ldsf[wave * VEC_DIM + koff];

    v8f acc0 = {0.f,0.f,0.f,0.f,0.f,0.f,0.f,0.f};
    v8f acc1 = {0.f,0.f,0.f,0.f,0.f,0.f,0.f,0.f};

    #pragma unroll
    for (int kc = 0; kc < VEC_DIM; kc += 8) {
        v2f a0, b0, a1, b1;
        a0.x = aptr[0];
        a0.y = aptr[NWORDS];
        b0.x = bbase[kc];
        b0.y = bbase[kc + 1];
        a1.x = aptr[4 * NWORDS];
        a1.y = aptr[5 * NWORDS];
        b1.x = bbase[kc + 4];
        b1.y = bbase[kc + 5];
        acc0 = __builtin_amdgcn_wmma_f32_16x16x4_f32(false, a0, false, b0,
                                                     (short)0, acc0, false, false);
        acc1 = __builtin_amdgcn_wmma_f32_16x16x4_f32(false, a1, false, b1,
                                                     (short)0, acc1, false, false);
        aptr += 8 * NWORDS;
    }

    // Combine the two accumulation chains (same D layout, elementwise add is valid).
    v8f acc = acc0 + acc1;

    // ---------------- Store: D[m,0] -> res[b, m] ----------------
    // Lane 0 holds D[0..7, 0] in acc[0..7]; lane 16 holds D[8..15, 0] in acc[0..7].
    if (lane == 0) {
        #pragma unroll
        for (int r = 0; r < 8; ++r)
            res[b * SSAMP + r] = acc[r];
    } else if (lane == 16) {
        #pragma unroll
        for (int r = 0; r < SSAMP - 8; ++r)
            res[b * SSAMP + 8 + r] = acc[r];
    }
}

extern "C" void kernel_launch(void* const* d_in, const int* in_sizes, int n_in,
                              void* d_out, int out_size, void* d_ws, size_t ws_size,
                              hipStream_t stream) {
    const int*   doc_ids     = (const int*)d_in[0];
    const int*   context_ids = (const int*)d_in[1];
    const int*   sample_ids  = (const int*)d_in[2];
    const float* P           = (const float*)d_in[3];
    const float* Wm          = (const float*)d_in[4];
    const float* Out         = (const float*)d_in[5];
    float*       res         = (float*)d_out;

    const int Btot   = in_sizes[0];                       // B = 16384
    const int blocks = (Btot + WAVES_PER_BLOCK - 1) / WAVES_PER_BLOCK;

    hipLaunchKernelGGL(pvdm_wmma_kernel, dim3(blocks), dim3(256), 0, stream,
                       doc_ids, context_ids, sample_ids, P, Wm, Out, res, Btot);
}